// Attention_68968584839415
// MI455X (gfx1250) — compile-verified
//
#include <hip/hip_runtime.h>
#include <cstdint>
#include <cstddef>

// Problem shape (fixed by the reference).
#define B_  4
#define LQ_ 4096
#define LK_ 4096
#define D_  128

typedef float v2f __attribute__((ext_vector_type(2)));
typedef float v4f __attribute__((ext_vector_type(4)));
typedef float v8f __attribute__((ext_vector_type(8)));

// ---------------------------------------------------------------------------
// Kernel 1: p[b,j] = softmax_j( key[b,j,:] . w_k )
// One 256-thread (8-wave) block per batch; each thread owns 16 rows of key.
// score = sq[i] + sk[j]  =>  softmax over j is independent of sq (shift
// invariance), so qry / w_q never need to be read.
// ---------------------------------------------------------------------------
__global__ __launch_bounds__(256) void k_softmax_keys(
    const float* __restrict__ key, const float* __restrict__ w_k,
    float* __restrict__ p)
{
  const int b   = blockIdx.x;
  const int tid = threadIdx.x;
  __shared__ v4f   wks[D_ / 4];
  __shared__ float red[256];

  if (tid < D_ / 4) wks[tid] = ((const v4f*)w_k)[tid];
  __syncthreads();

  const float* kb = key + (size_t)b * LK_ * D_;
  float sk[LK_ / 256];
  float lmax = -3.402823466e+38f;
  #pragma unroll 4
  for (int r = 0; r < LK_ / 256; ++r) {
    const int  j   = r * 256 + tid;
    const v4f* row = (const v4f*)(kb + (size_t)j * D_);
    float acc = 0.f;
    #pragma unroll
    for (int c = 0; c < D_ / 4; ++c) {
      v4f kv = row[c];
      v4f wv = wks[c];
      acc = fmaf(kv.x, wv.x, acc);
      acc = fmaf(kv.y, wv.y, acc);
      acc = fmaf(kv.z, wv.z, acc);
      acc = fmaf(kv.w, wv.w, acc);
    }
    sk[r] = acc;
    lmax  = fmaxf(lmax, acc);
  }

  // block-wide max
  red[tid] = lmax; __syncthreads();
  for (int s = 128; s > 0; s >>= 1) {
    if (tid < s) red[tid] = fmaxf(red[tid], red[tid + s]);
    __syncthreads();
  }
  const float m = red[0];
  __syncthreads();

  // exp + block-wide sum
  float lsum = 0.f;
  #pragma unroll
  for (int r = 0; r < LK_ / 256; ++r) { sk[r] = __expf(sk[r] - m); lsum += sk[r]; }
  red[tid] = lsum; __syncthreads();
  for (int s = 128; s > 0; s >>= 1) {
    if (tid < s) red[tid] += red[tid + s];
    __syncthreads();
  }
  const float inv = 1.0f / red[0];

  float* pb = p + (size_t)b * LK_;
  #pragma unroll
  for (int r = 0; r < LK_ / 256; ++r) pb[r * 256 + tid] = sk[r] * inv;
}

// ---------------------------------------------------------------------------
// Kernel 2: o[b,d] = sum_j p[b,j] * val[b,j,d]   via V_WMMA_F32_16X16X4_F32.
// One wave per (batch, 16-wide d tile): 4*8 = 32 waves.
// A[m,k] = p[j0+k]  (broadcast over m)  -> every output row of D equals o.
// B[k,n] = val[j0+k, d0+n].
// A and B use the SAME k -> (lane-half, vgpr) placement (per the documented
// 16x4 f32 A layout), so the k-reduction is invariant to that mapping.
// 4 independent accumulator chains (j, j+4, j+8, j+12) hide WMMA latency.
// ---------------------------------------------------------------------------
__global__ __launch_bounds__(32) void k_pv_wmma(
    const float* __restrict__ val, const float* __restrict__ p,
    float* __restrict__ o)
{
  const int b    = blockIdx.x >> 3;          // D_/16 = 8 tiles per batch
  const int d0   = (blockIdx.x & 7) * 16;
  const int lane = threadIdx.x;              // wave32
  const int n    = lane & 15;
  const int kg   = (lane >> 4) * 2;          // k base for this half-wave: 0 or 2

  const float* vb = val + (size_t)b * LK_ * D_ + d0 + n;
  const float* pb = p + (size_t)b * LK_;

  v8f acc[4] = {v8f{}, v8f{}, v8f{}, v8f{}};
  for (int j = 0; j < LK_; j += 16) {
    #pragma unroll
    for (int u = 0; u < 4; ++u) {
      const int j0 = j + 4 * u;
      v2f a, bm;
      a.x  = pb[j0 + kg];
      a.y  = pb[j0 + kg + 1];
      bm.x = vb[(size_t)(j0 + kg) * D_];
      bm.y = vb[(size_t)(j0 + kg + 1) * D_];
      acc[u] = __builtin_amdgcn_wmma_f32_16x16x4_f32(
          false, a, false, bm, (short)0, acc[u], false, false);
    }
  }
  v8f r = (acc[0] + acc[1]) + (acc[2] + acc[3]);
  // C/D layout: lanes 0..15 hold N=lane, VGPR0 = row M=0; all rows equal o.
  if (lane < 16) o[(size_t)b * D_ + d0 + lane] = r[0];
}

// ---------------------------------------------------------------------------
// Kernel 3: broadcast.  att[b,i,:] = p[b,:]  and  out[b,i,:] = o[b,:].
// Grid (LQ/32, B); each 256-thread block owns 32 att rows of one batch.
// p[b,:] (16 KB) is staged into LDS with gfx1250 async global->LDS B128 DMA
// (ASYNCcnt), then streamed out with non-temporal B128 stores (att > L2).
// ---------------------------------------------------------------------------
__global__ __launch_bounds__(256) void k_broadcast(
    const float* __restrict__ p, const float* __restrict__ o,
    float* __restrict__ out, float* __restrict__ att)
{
  const int b   = blockIdx.y;
  const int tid = threadIdx.x;
  __shared__ float ps[LK_];   // 16 KB
  __shared__ float os[D_];

  {
    const float* src = p + (size_t)b * LK_;
    #pragma unroll
    for (int it = 0; it < 4; ++it) {
      const int e = (it * 256 + tid) * 4;                    // element index
      unsigned lds_addr = (unsigned)(uintptr_t)(&ps[e]);     // low 32 bits of a
                                                             // generic LDS ptr
                                                             // == LDS offset
      unsigned voff = (unsigned)(e * sizeof(float));         // byte offset
      asm volatile("global_load_async_to_lds_b128 %0, %1, %2"
                   :: "v"(lds_addr), "v"(voff), "s"(src)
                   : "memory");
    }
  }
  if (tid < D_) os[tid] = o[(size_t)b * D_ + tid];
  asm volatile("s_wait_asynccnt 0" ::: "memory");
  __syncthreads();

  // Hoist this thread's slice of p (4 x 16B) and o into registers.
  v4f pr[4];
  #pragma unroll
  for (int c = 0; c < 4; ++c) pr[c] = *(const v4f*)&ps[(c * 256 + tid) * 4];
  v4f orv = {};
  if (tid < D_ / 4) orv = *(const v4f*)&os[tid * 4];

  float* attb = att + (size_t)b * LQ_ * LK_;
  float* outb = out + (size_t)b * LQ_ * D_;
  const size_t row0 = (size_t)blockIdx.x * 32;
  for (int r = 0; r < 32; ++r) {
    const size_t row  = row0 + r;
    float*       arow = attb + row * (size_t)LK_;
    #pragma unroll
    for (int c = 0; c < 4; ++c)
      __builtin_nontemporal_store(pr[c], (v4f*)(arow + (c * 256 + tid) * 4));
    if (tid < D_ / 4)
      __builtin_nontemporal_store(orv, (v4f*)(outb + row * D_ + tid * 4));
  }
}

// ---------------------------------------------------------------------------
extern "C" void kernel_launch(void* const* d_in, const int* in_sizes, int n_in,
                              void* d_out, int out_size, void* d_ws, size_t ws_size,
                              hipStream_t stream)
{
  (void)in_sizes; (void)n_in; (void)out_size; (void)ws_size;
  // inputs: 0=qry (unused: softmax shift-invariance), 1=key, 2=val,
  //         3=w_q (unused), 4=w_k
  const float* key = (const float*)d_in[1];
  const float* val = (const float*)d_in[2];
  const float* w_k = (const float*)d_in[4];

  float* out = (float*)d_out;                          // [B, LQ, D]
  float* att = out + (size_t)B_ * LQ_ * D_;            // [B, LQ, LK]

  float* p = (float*)d_ws;                             // [B, LK]
  float* o = p + (size_t)B_ * LK_;                     // [B, D]

  k_softmax_keys<<<dim3(B_), dim3(256), 0, stream>>>(key, w_k, p);
  k_pv_wmma<<<dim3(B_ * (D_ / 16)), dim3(32), 0, stream>>>(val, p, o);
  k_broadcast<<<dim3(LQ_ / 32, B_), dim3(256), 0, stream>>>(p, o, out, att);
}